// MultiscaleMix_43112881717766
// MI455X (gfx1250) — compile-verified
//
#include <hip/hip_runtime.h>
#include <hip/hip_bf16.h>

// MI455X / gfx1250, wave32. FP32 GEMMs via V_WMMA_F32_16X16X4_F32 with
// double-buffered LDS staging through GLOBAL_LOAD_ASYNC_TO_LDS_B128
// (ASYNCcnt-tracked, in-order completion -> s_wait_asynccnt 4 pipelining).

typedef float v2f __attribute__((ext_vector_type(2)));
typedef float v8f __attribute__((ext_vector_type(8)));

#define MBLK 64
#define NBLK 64
#define KC   32
#define APITCH 36   // floats; 144B rows -> 16B aligned, conflict-free b64 reads
#define BPITCH 72   // floats; 288B rows -> 16B aligned, half-wave groups 16 banks apart

__device__ __forceinline__ void async_b128_to_lds(const float* gaddr, unsigned lds_byte_off) {
    // GV mode: VDST = per-lane LDS byte address, VADDR = 64-bit global address.
    asm volatile("global_load_async_to_lds_b128 %0, %1, off"
                 :: "v"(lds_byte_off), "v"(gaddr)
                 : "memory");
}
__device__ __forceinline__ void wait_async0() {
    asm volatile("s_wait_asynccnt 0x0" ::: "memory");
}
__device__ __forceinline__ void wait_async4() {
    // <=4 outstanding: current chunk's 4 transfers done, next chunk's 4 in flight.
    asm volatile("s_wait_asynccnt 0x4" ::: "memory");
}

// Batched GEMM: Y[b] (MxN) = W (MxK, row-major) * X[b] (KxN, row-major)
//               [+bias per row][ReLU][accumulate into Y]
// Block = 256 threads = 8 waves. Block tile 64x64, K chunk 32, double-buffered LDS.
// Wave w: n-tile = (w&3)*16, m-tiles = (w>>2)*32 and (w>>2)*32+16 (B fragment reused).
// Requires: M % 64 == 0, N % 64 == 0, K % 32 == 0.
__global__ void __launch_bounds__(256)
gemm_wmma_f32(const float* __restrict__ W, const float* __restrict__ X,
              const float* __restrict__ bias, float* __restrict__ Y,
              int M, int N, int K, long long strideX, long long strideY,
              int relu, int accum)
{
    __shared__ float As[2][MBLK * APITCH];  // [2][64][36]
    __shared__ float Bs[2][KC * BPITCH];    // [2][32][72]

    const int tid   = threadIdx.x;
    const int lane  = tid & 31;
    const int wave  = tid >> 5;
    const int lcol  = lane & 15;
    const int lhalf = lane >> 4;         // 0: K+0/K+1, rows r ; 1: K+2/K+3, rows r+8

    const int m0 = blockIdx.y * MBLK;
    const int n0 = blockIdx.x * NBLK;
    const int mA0 = (wave >> 2) * 32;              // first of the wave's two 16-row tiles
    const int ncol = (wave & 3) * 16 + lcol;       // column within block tile
    const int gcol = n0 + ncol;                    // global output column of this lane

    const float* __restrict__ Xb = X + (long long)blockIdx.z * strideX;
    float* __restrict__ Yb       = Y + (long long)blockIdx.z * strideY;

    // ---- accumulator init (C/D layout: VGPR r -> rows base+r (lanes 0-15) / base+8+r (16-31))
    v8f acc0, acc1;
#pragma unroll
    for (int r = 0; r < 8; ++r) {
        const int row0 = m0 + mA0 + r + 8 * lhalf;
        const int row1 = row0 + 16;
        float v0, v1;
        if (accum) {
            v0 = Yb[(long long)row0 * N + gcol];
            v1 = Yb[(long long)row1 * N + gcol];
        } else if (bias) {
            v0 = bias[row0];
            v1 = bias[row1];
        } else {
            v0 = 0.0f; v1 = 0.0f;
        }
        acc0[r] = v0; acc1[r] = v1;
    }

    // ---- cooperative-load index precompute
    // A tile: 64 rows x 32 cols. thread t -> rows (t>>3, t>>3 + 32), col (t&7)*4.
    const int arow = tid >> 3;
    const int acol = (tid & 7) * 4;
    // B tile: 32 rows x 64 cols. thread t -> rows (t>>4, t>>4 + 16), col (t&15)*4.
    const int brow = tid >> 4;
    const int bcol = (tid & 15) * 4;

    unsigned aoff0[2], aoff1[2], boff0[2], boff1[2];
#pragma unroll
    for (int buf = 0; buf < 2; ++buf) {
        aoff0[buf] = (unsigned)(size_t)&As[buf][arow * APITCH + acol];
        aoff1[buf] = (unsigned)(size_t)&As[buf][(arow + 32) * APITCH + acol];
        boff0[buf] = (unsigned)(size_t)&Bs[buf][brow * BPITCH + bcol];
        boff1[buf] = (unsigned)(size_t)&Bs[buf][(brow + 16) * BPITCH + bcol];
    }

    const float* __restrict__ Wa0 = W + (long long)(m0 + arow) * K + acol;
    const float* __restrict__ Wa1 = W + (long long)(m0 + arow + 32) * K + acol;
    const float* __restrict__ Xb0 = Xb + (long long)brow * N + n0 + bcol;
    const float* __restrict__ Xb1 = Xb + (long long)(brow + 16) * N + n0 + bcol;

    auto issue_chunk = [&](int buf, int k) {
        async_b128_to_lds(Wa0 + k, aoff0[buf]);
        async_b128_to_lds(Wa1 + k, aoff1[buf]);
        async_b128_to_lds(Xb0 + (long long)k * N, boff0[buf]);
        async_b128_to_lds(Xb1 + (long long)k * N, boff1[buf]);
    };

    const int nchunks = K / KC;
    issue_chunk(0, 0);

    for (int c = 0; c < nchunks; ++c) {
        const int cur = c & 1;
        if (c + 1 < nchunks) {
            issue_chunk(cur ^ 1, (c + 1) * KC);  // prefetch next chunk into other buffer
            wait_async4();                        // current chunk's 4 transfers complete
        } else {
            wait_async0();
        }
        __syncthreads();

        const float* __restrict__ Ab = &As[cur][0];
        const float* __restrict__ Bb = &Bs[cur][0];

        // ---- 8 x (2 WMMA) over the K chunk
#pragma unroll
        for (int kk = 0; kk < KC; kk += 4) {
            v2f a0 = *(const v2f*)&Ab[(mA0 + lcol) * APITCH + kk + 2 * lhalf];
            v2f a1 = *(const v2f*)&Ab[(mA0 + 16 + lcol) * APITCH + kk + 2 * lhalf];
            v2f b;
            b.x = Bb[(kk + 2 * lhalf) * BPITCH + ncol];
            b.y = Bb[(kk + 2 * lhalf + 1) * BPITCH + ncol];
            acc0 = __builtin_amdgcn_wmma_f32_16x16x4_f32(false, a0, false, b,
                                                         (short)0, acc0, false, false);
            acc1 = __builtin_amdgcn_wmma_f32_16x16x4_f32(false, a1, false, b,
                                                         (short)0, acc1, false, false);
        }
        __syncthreads();  // protect buffer 'cur' before chunk c+2 overwrites it
    }

    if (relu) {
#pragma unroll
        for (int r = 0; r < 8; ++r) {
            acc0[r] = fmaxf(acc0[r], 0.0f);
            acc1[r] = fmaxf(acc1[r], 0.0f);
        }
    }
#pragma unroll
    for (int r = 0; r < 8; ++r) {
        const int row0 = m0 + mA0 + r + 8 * lhalf;
        Yb[(long long)row0 * N + gcol]        = acc0[r];
        Yb[(long long)(row0 + 16) * N + gcol] = acc1[r];
    }
}

// dec_out init: dec[b,p,c] = sum_i predict_b[i][p]
__global__ void dec_init_kernel(const float* __restrict__ b0, const float* __restrict__ b1,
                                const float* __restrict__ b2, const float* __restrict__ b3,
                                float* __restrict__ dec, long long total, int Cc)
{
    long long idx = (long long)blockIdx.x * blockDim.x + threadIdx.x;
    if (idx >= total) return;
    int p = (int)((idx / Cc) & 511);  // TS_LEN = 512
    dec[idx] = b0[p] + b1[p] + b2[p] + b3[p];
}

// Per (b,c): 4x4 Gram over t of P[m][b][t][c]; normalize by per-(b,c) time-norms; atomically
// accumulate S[m][n][b] += d_mn / (||P_m|| * ||P_n||).  P layout: [m][b][t][c] (stride B*T*C).
__global__ void gram_scale_kernel(const float* __restrict__ P, int Tlen, int Cc, int Bb,
                                  float* __restrict__ S)
{
    int idx = blockIdx.x * blockDim.x + threadIdx.x;  // = b*Cc + c
    if (idx >= Bb * Cc) return;
    const int b = idx / Cc;
    const int c = idx - b * Cc;
    const long long mstride = (long long)Bb * Tlen * Cc;
    const float* __restrict__ p0 = P + (long long)b * Tlen * Cc + c;

    float d[4][4];
#pragma unroll
    for (int m = 0; m < 4; ++m)
#pragma unroll
        for (int n = 0; n < 4; ++n) d[m][n] = 0.0f;

    for (int t = 0; t < Tlen; ++t) {
        float v[4];
#pragma unroll
        for (int m = 0; m < 4; ++m)
            v[m] = p0[m * mstride + (long long)t * Cc];
#pragma unroll
        for (int m = 0; m < 4; ++m)
#pragma unroll
            for (int n = m; n < 4; ++n) d[m][n] += v[m] * v[n];
    }
    float nrm[4];
#pragma unroll
    for (int m = 0; m < 4; ++m) nrm[m] = fmaxf(sqrtf(d[m][m]), 1e-12f);
#pragma unroll
    for (int m = 0; m < 4; ++m)
#pragma unroll
        for (int n = 0; n < 4; ++n) {
            float dv = (n >= m) ? d[m][n] : d[n][m];
            atomicAdd(&S[(m * 4 + n) * Bb + b], dv / (nrm[m] * nrm[n]));
        }
}

// L = (1/(2*B*3)) * sum_{i=0..2, b} [ -log(E[i,i+1]/denom_i) - log(E[i+1,i]/denom_{i+1}) ]
__global__ void loss_kernel(const float* __restrict__ S, float* __restrict__ out, int Bb)
{
    __shared__ float red[64];
    const int b = threadIdx.x;
    float acc = 0.0f;
    if (b < Bb) {
        for (int i = 0; i < 3; ++i) {
            const float* Si = S + i * 16 * Bb;
            float E[4][4];
#pragma unroll
            for (int m = 0; m < 4; ++m)
#pragma unroll
                for (int n = 0; n < 4; ++n)
                    E[m][n] = expf(Si[(m * 4 + n) * Bb + b] * 5.0f);  // /0.2
            float di = 0.0f, di1 = 0.0f;
#pragma unroll
            for (int n = 0; n < 4; ++n) { di += E[i][n]; di1 += E[i + 1][n]; }
            di  -= E[i][i];
            di1 -= E[i + 1][i + 1];
            acc += -logf(E[i][i + 1] / di) - logf(E[i + 1][i] / di1);
        }
    }
    red[threadIdx.x] = acc;
    __syncthreads();
    for (int s = 32; s > 0; s >>= 1) {
        if (threadIdx.x < s) red[threadIdx.x] += red[threadIdx.x + s];
        __syncthreads();
    }
    if (threadIdx.x == 0) out[0] = red[0] / (2.0f * Bb * 3.0f);
}

extern "C" void kernel_launch(void* const* d_in, const int* in_sizes, int n_in,
                              void* d_out, int out_size, void* d_ws, size_t ws_size,
                              hipStream_t stream)
{
    (void)in_sizes; (void)n_in; (void)out_size; (void)ws_size;
    const int Bb = 64, Cc = 512, TS = 512;
    const int Tarr[4] = {512, 256, 128, 64};

    const float* x[4];
    const float* predW[4];
    const float* predB[4];
    const float* fc1w[3][4]; const float* fc1b[3][4];
    const float* fc2w[3][4]; const float* fc2b[3][4];
    for (int i = 0; i < 4; ++i) {
        x[i]     = (const float*)d_in[i];
        predW[i] = (const float*)d_in[4 + i];
        predB[i] = (const float*)d_in[8 + i];
    }
    for (int i = 0; i < 3; ++i)
        for (int m = 0; m < 4; ++m) {
            fc1w[i][m] = (const float*)d_in[12 + i * 4 + m];
            fc1b[i][m] = (const float*)d_in[24 + i * 4 + m];
            fc2w[i][m] = (const float*)d_in[36 + i * 4 + m];
            fc2b[i][m] = (const float*)d_in[48 + i * 4 + m];
        }

    // Workspace carve-up (floats): P (4 scales at max T=512), H (one scale), S (3*16*B)
    float* Pbuf = (float*)d_ws;
    const size_t Pelems = 4ull * Bb * 512 * Cc;        // 67,108,864 floats
    float* Hbuf = Pbuf + Pelems;
    const size_t Helems = (size_t)Bb * 512 * Cc;       // 16,777,216 floats
    float* Sbuf = Hbuf + Helems;
    hipMemsetAsync(Sbuf, 0, 3 * 16 * Bb * sizeof(float), stream);

    // ---- Contrastive branch: projections + Gram per level i ----
    for (int i = 0; i < 3; ++i) {
        const int Ti = Tarr[i];
        const long long pms = (long long)Bb * Ti * Cc;  // per-m stride in Pbuf
        dim3 grid(Cc / NBLK, Ti / MBLK, Bb);
        for (int m = 0; m < 4; ++m) {
            const int Tm = Tarr[m];
            // H = relu(W1 @ x_m + b1)   (M=Ti, N=C, K=Tm)
            gemm_wmma_f32<<<grid, 256, 0, stream>>>(
                fc1w[i][m], x[m], fc1b[i][m], Hbuf,
                Ti, Cc, Tm, (long long)Tm * Cc, (long long)Ti * Cc, /*relu=*/1, /*accum=*/0);
            // P_m = W2 @ H + b2         (M=Ti, N=C, K=Ti)
            gemm_wmma_f32<<<grid, 256, 0, stream>>>(
                fc2w[i][m], Hbuf, fc2b[i][m], Pbuf + (size_t)m * pms,
                Ti, Cc, Ti, (long long)Ti * Cc, (long long)Ti * Cc, /*relu=*/0, /*accum=*/0);
        }
        const int bc = Bb * Cc;
        gram_scale_kernel<<<(bc + 255) / 256, 256, 0, stream>>>(
            Pbuf, Ti, Cc, Bb, Sbuf + i * 16 * Bb);
    }

    // ---- Decoder branch: dec = sum_i (predict_w[i] @ x_i + predict_b[i]) ----
    float* Lout = (float*)d_out;
    float* dec  = Lout + 1;
    const long long dtotal = (long long)Bb * TS * Cc;
    dec_init_kernel<<<(int)((dtotal + 255) / 256), 256, 0, stream>>>(
        predB[0], predB[1], predB[2], predB[3], dec, dtotal, Cc);
    for (int i = 0; i < 4; ++i) {
        dim3 grid(Cc / NBLK, TS / MBLK, Bb);
        gemm_wmma_f32<<<grid, 256, 0, stream>>>(
            predW[i], x[i], nullptr, dec,
            TS, Cc, Tarr[i], (long long)Tarr[i] * Cc, (long long)TS * Cc, /*relu=*/0, /*accum=*/1);
    }

    // ---- Loss reduction ----
    loss_kernel<<<1, 64, 0, stream>>>(Sbuf, Lout, Bb);
}